// SchNETInteractionBlock_76192719831220
// MI455X (gfx1250) — compile-verified
//
#include <hip/hip_runtime.h>
#include <hip/hip_bf16.h>
#include <math.h>

// SchNet interaction block for MI455X (gfx1250, wave32).
//  K1: h   = x @ W_in + b_in                  (V_WMMA_F32_16X16X4_F32)
//  K2: agg = segment_sum(h[idx_j] * ssp(f_ij@W_f+b_f)*rcut, idx_i)
//      (fused filter-MLP in regs, wave-per-pair, f32 atomics into L2-resident agg)
//  K3: out = ssp(agg @ W_out + b_out)         (V_WMMA_F32_16X16X4_F32)

typedef __attribute__((ext_vector_type(2))) float v2f;
typedef __attribute__((ext_vector_type(8))) float v8f;

#define DCH   128   // D_BASIS == D_FILT
#define NRBF  20

__device__ __forceinline__ float shifted_softplus(float v) {
  float sp = (v > 20.0f) ? v : log1pf(__expf(v));
  return sp - 0.6931471805599453f;
}

// ---------------------------------------------------------------------------
// Y[M,128] = act(X[M,128] @ W[128,128] + bias[128])
// Block = 256 threads = 8 waves. Block owns 16 rows; wave w owns cols [16w,16w+16).
// fp32 WMMA 16x16x4, K swept 0..128 in steps of 4.
// A frag (16x4 f32, 2 VGPR): lanes 0-15 hold {K,K+1}, lanes 16-31 hold {K+2,K+3}.
// B frag (4x16 f32, 2 VGPR): v0 = rows K (lo lanes) / K+1 (hi lanes), v1 = K+2 / K+3.
// C/D (16x16 f32, 8 VGPR): vr = row r (lo lanes) / row r+8 (hi lanes).
// ---------------------------------------------------------------------------
template <bool APPLY_SSP>
__global__ __launch_bounds__(256) void gemm128_wmma(const float* __restrict__ X,
                                                    const float* __restrict__ W,
                                                    const float* __restrict__ bias,
                                                    float* __restrict__ Y, int M) {
  // 16 rows x 128 cols, pad row stride to 132 floats: 8B-aligned b64 loads and
  // conflict-free column access (stride 132 % 64 banks = 4).
  __shared__ float xs[16 * 132];

  const int tid = threadIdx.x;
  const int m0  = blockIdx.x * 16;

  for (int i = tid; i < 16 * DCH; i += 256) {
    int r = i >> 7, c = i & 127;
    int m = m0 + r;
    xs[r * 132 + c] = (m < M) ? X[(size_t)m * DCH + c] : 0.0f;
  }
  __syncthreads();

  const int wave = tid >> 5;
  const int lane = tid & 31;
  const int hi   = lane >> 4;    // 0: lanes 0-15, 1: lanes 16-31
  const int l15  = lane & 15;
  const int n    = wave * 16 + l15;

  v8f acc = {};
#pragma unroll
  for (int k = 0; k < DCH; k += 4) {
    v2f a = *(const v2f*)&xs[l15 * 132 + k + 2 * hi];
    v2f b;
    b.x = W[(k + hi) * DCH + n];
    b.y = W[(k + 2 + hi) * DCH + n];
    acc = __builtin_amdgcn_wmma_f32_16x16x4_f32(false, a, false, b, (short)0, acc,
                                                false, false);
  }

  const float bn = bias[n];
#pragma unroll
  for (int r = 0; r < 8; ++r) {
    int m = m0 + r + 8 * hi;
    if (m < M) {
      float v = acc[r] + bn;
      if (APPLY_SSP) v = shifted_softplus(v);
      Y[(size_t)m * DCH + n] = v;
    }
  }
}

// ---------------------------------------------------------------------------
// One wave32 per pair; lane owns 4 contiguous channels (float4 gather, 4 atomics).
// Filter weights (20x128) + bias staged in LDS once per block (grid-stride loop
// amortizes the staging). f_ij reads are wave-uniform -> broadcast loads.
// ---------------------------------------------------------------------------
__global__ __launch_bounds__(256) void pair_scatter(const float* __restrict__ h,
                                                    const int* __restrict__ pairlist,
                                                    const float* __restrict__ f_ij,
                                                    const float* __restrict__ rcut,
                                                    const float* __restrict__ W_f,
                                                    const float* __restrict__ b_f,
                                                    float* __restrict__ agg,
                                                    int n_pairs) {
  __shared__ float wf[NRBF * DCH];
  __shared__ float bf[DCH];
  const int tid = threadIdx.x;
  for (int i = tid; i < NRBF * DCH; i += 256) wf[i] = W_f[i];
  if (tid < DCH) bf[tid] = b_f[tid];
  __syncthreads();

  const int lane = tid & 31;
  const int wave = tid >> 5;
  const int c0   = lane * 4;
  const int stride = gridDim.x * 8;   // waves in grid

  for (int p = blockIdx.x * 8 + wave; p < n_pairs; p += stride) {
    // filter MLP: ssp(f_ij[p] @ W_f + b_f) * rcut[p], 4 channels per lane
    float a0 = bf[c0], a1 = bf[c0 + 1], a2 = bf[c0 + 2], a3 = bf[c0 + 3];
#pragma unroll
    for (int r = 0; r < NRBF; ++r) {
      float fv = f_ij[(size_t)p * NRBF + r];   // uniform across wave -> broadcast
      const float* wr = &wf[r * DCH + c0];
      a0 = fmaf(fv, wr[0], a0);
      a1 = fmaf(fv, wr[1], a1);
      a2 = fmaf(fv, wr[2], a2);
      a3 = fmaf(fv, wr[3], a3);
    }
    const float rc = rcut[p];
    a0 = shifted_softplus(a0) * rc;
    a1 = shifted_softplus(a1) * rc;
    a2 = shifted_softplus(a2) * rc;
    a3 = shifted_softplus(a3) * rc;

    const int idx_i = pairlist[p];
    const int idx_j = pairlist[n_pairs + p];

    const float4 hj = *(const float4*)&h[(size_t)idx_j * DCH + c0];
    float* dst = &agg[(size_t)idx_i * DCH + c0];
    unsafeAtomicAdd(dst + 0, hj.x * a0);   // global_atomic_add_f32 (agg is L2-resident)
    unsafeAtomicAdd(dst + 1, hj.y * a1);
    unsafeAtomicAdd(dst + 2, hj.z * a2);
    unsafeAtomicAdd(dst + 3, hj.w * a3);
  }
}

__global__ void zero_f32(float* __restrict__ p, int n) {
  int i = blockIdx.x * blockDim.x + threadIdx.x;
  if (i < n) p[i] = 0.0f;
}

extern "C" void kernel_launch(void* const* d_in, const int* in_sizes, int n_in,
                              void* d_out, int out_size, void* d_ws, size_t ws_size,
                              hipStream_t stream) {
  const float* x     = (const float*)d_in[0];
  const int*   pl    = (const int*)  d_in[1];   // pairlist [2, n_pairs]
  const float* f_ij  = (const float*)d_in[2];
  const float* rcut  = (const float*)d_in[3];
  const float* W_in  = (const float*)d_in[4];
  const float* b_in  = (const float*)d_in[5];
  const float* W_f   = (const float*)d_in[6];
  const float* b_f   = (const float*)d_in[7];
  const float* W_out = (const float*)d_in[8];
  const float* b_out = (const float*)d_in[9];
  float* out = (float*)d_out;

  const int n_atoms = in_sizes[0] / DCH;
  const int n_pairs = in_sizes[3];

  float* h   = (float*)d_ws;                       // [n_atoms, 128]
  float* agg = h + (size_t)n_atoms * DCH;          // [n_atoms, 128]

  const dim3 blk(256);
  const int mtiles = (n_atoms + 15) / 16;

  // K1: h = x @ W_in + b_in
  gemm128_wmma<false><<<mtiles, blk, 0, stream>>>(x, W_in, b_in, h, n_atoms);

  // zero the accumulator every call (deterministic under graph replay)
  const int nz = n_atoms * DCH;
  zero_f32<<<(nz + 255) / 256, blk, 0, stream>>>(agg, nz);

  // K2: fused filter + gather + scatter-add
  pair_scatter<<<4096, blk, 0, stream>>>(h, pl, f_ij, rcut, W_f, b_f, agg, n_pairs);

  // K3: out = ssp(agg @ W_out + b_out)
  gemm128_wmma<true><<<mtiles, blk, 0, stream>>>(agg, W_out, b_out, out, n_atoms);
}